// GroupMamba_9405978378817
// MI455X (gfx1250) — compile-verified
//
#include <hip/hip_runtime.h>
#include <math.h>
#include <stdint.h>

// Problem constants (match reference)
#define Bq   4
#define Lq   2048
#define Dq   512
#define Gq   4
#define GSq  128
#define DIq  256
#define DSq  16
#define DCq  4
#define DTRq 8
#define XDBL 40                 // DTR + 2*DS
#define BL   (Bq * Lq)          // 8192 rows per group
#define ROWS ((size_t)Gq * BL)  // 32768 total (g,b,l) rows

typedef __attribute__((ext_vector_type(2))) float v2f;
typedef __attribute__((ext_vector_type(8))) float v8f;

static __device__ __forceinline__ v8f wmma4(v2f a, v2f b, v8f c) {
    // D = A(16x4 f32) * B(4x16 f32) + C(16x16 f32)
    return __builtin_amdgcn_wmma_f32_16x16x4_f32(false, a, false, b,
                                                 (short)0, c, false, false);
}

static __device__ __forceinline__ float silu_f(float x) {
    return x / (1.0f + __expf(-x));
}

// ---------------------------------------------------------------------------
// Kernel 1: in-projection. Per group g: xz = x_g[8192x128] * W_in[g]^T[128x512]
// One wave computes a 16(M) x 64(N) strip: A fragment reused across 4 WMMAs.
// cols [0,256) -> xc, cols [256,512) -> z.
// ---------------------------------------------------------------------------
__global__ __launch_bounds__(256) void inproj_kernel(const float* __restrict__ x,
                                                     const float* __restrict__ W_in,
                                                     float* __restrict__ xc,
                                                     float* __restrict__ zb) {
    const int Mt = BL / 16;          // 512 M-tiles
    const int NW = (2 * DIq) / 64;   // 8 N-strips of 64
    int wave = (blockIdx.x * blockDim.x + threadIdx.x) >> 5;
    int lane = threadIdx.x & 31;
    int nw = wave % NW;
    int mt = (wave / NW) % Mt;
    int g  = wave / (NW * Mt);
    if (g >= Gq) return;

    int mh = lane >> 4;      // k-pair selector / M row offset 8
    int mn = lane & 15;      // A row within tile == B column within tile

    const float* arow = x + (size_t)(mt * 16 + mn) * Dq + g * GSq;  // x[b,l, g*128 + k]
    const float* br[4];
#pragma unroll
    for (int j = 0; j < 4; ++j)                                     // W_in[g, n, k]
        br[j] = W_in + ((size_t)g * (2 * DIq) + (nw * 64 + j * 16 + mn)) * GSq;

    v8f acc0 = {0.f,0.f,0.f,0.f,0.f,0.f,0.f,0.f};
    v8f acc1 = acc0, acc2 = acc0, acc3 = acc0;
#pragma unroll 4
    for (int k0 = 0; k0 < GSq; k0 += 4) {
        int kb = k0 + 2 * mh;                          // even -> 8B aligned
        v2f a  = *(const v2f*)(arow + kb);
        v2f b0 = *(const v2f*)(br[0] + kb);
        v2f b1 = *(const v2f*)(br[1] + kb);
        v2f b2 = *(const v2f*)(br[2] + kb);
        v2f b3 = *(const v2f*)(br[3] + kb);
        acc0 = wmma4(a, b0, acc0);
        acc1 = wmma4(a, b1, acc1);
        acc2 = wmma4(a, b2, acc2);
        acc3 = wmma4(a, b3, acc3);
    }
    v8f accs[4] = {acc0, acc1, acc2, acc3};
#pragma unroll
    for (int j = 0; j < 4; ++j) {
        int e = nw * 64 + j * 16 + mn;
#pragma unroll
        for (int r = 0; r < 8; ++r) {
            int orow = mt * 16 + r + 8 * mh;
            float v  = accs[j][r];
            if (e < DIq) xc[((size_t)g * BL + orow) * DIq + e]         = v;
            else         zb[((size_t)g * BL + orow) * DIq + (e - DIq)] = v;
        }
    }
}

// ---------------------------------------------------------------------------
// Kernel 2: depthwise causal conv (DC=4) + bias + SiLU  -> u
// ---------------------------------------------------------------------------
__global__ __launch_bounds__(256) void conv_silu_kernel(const float* __restrict__ xc,
                                                        const float* __restrict__ conv_w,
                                                        const float* __restrict__ conv_b,
                                                        float* __restrict__ u) {
    size_t t = (size_t)blockIdx.x * blockDim.x + threadIdx.x;
    if (t >= ROWS * DIq) return;
    int    d    = (int)(t % DIq);
    size_t rowg = t / DIq;               // g*BL + b*L + l
    int    l    = (int)(rowg % Lq);
    int    g    = (int)(rowg / BL);
    int    gd   = g * DIq + d;

    float acc = conv_b[gd];
    const float* w = conv_w + (size_t)gd * DCq;
    size_t base0 = (rowg - (size_t)l) * DIq + d;   // row at l = 0
#pragma unroll
    for (int k = 0; k < DCq; ++k) {
        int ll = l - (DCq - 1) + k;
        if (ll >= 0) acc += w[k] * xc[base0 + (size_t)ll * DIq];
    }
    u[t] = silu_f(acc);
}

// ---------------------------------------------------------------------------
// Kernel 3: x_dbl = u_row[256] . W_x[g, j, :]   (40 outputs per row)
// ---------------------------------------------------------------------------
__global__ __launch_bounds__(256) void xdbl_kernel(const float* __restrict__ u,
                                                   const float* __restrict__ W_x,
                                                   float* __restrict__ xdbl) {
    size_t t = (size_t)blockIdx.x * blockDim.x + threadIdx.x;
    if (t >= ROWS * XDBL) return;
    int    j   = (int)(t % XDBL);
    size_t row = t / XDBL;
    int    g   = (int)(row / BL);
    const float* ur = u   + row * DIq;
    const float* wr = W_x + ((size_t)g * XDBL + j) * DIq;
    float s = 0.f;
#pragma unroll 4
    for (int k = 0; k < DIq; k += 4) {
        s += ur[k] * wr[k] + ur[k + 1] * wr[k + 1] +
             ur[k + 2] * wr[k + 2] + ur[k + 3] * wr[k + 3];
    }
    xdbl[t] = s;
}

// ---------------------------------------------------------------------------
// Kernel 4: dt = softplus(dt_in[8] . W_dt[g,d,:] + b_dt[g,d])
// ---------------------------------------------------------------------------
__global__ __launch_bounds__(256) void dt_kernel(const float* __restrict__ xdbl,
                                                 const float* __restrict__ W_dt,
                                                 const float* __restrict__ b_dt,
                                                 float* __restrict__ dt) {
    size_t t = (size_t)blockIdx.x * blockDim.x + threadIdx.x;
    if (t >= ROWS * DIq) return;
    int    d   = (int)(t % DIq);
    size_t row = t / DIq;
    int    g   = (int)(row / BL);
    const float* xr = xdbl + row * XDBL;                      // dt_in = xr[0..7]
    const float* wr = W_dt + ((size_t)g * DIq + d) * DTRq;
    float s = b_dt[g * DIq + d];
#pragma unroll
    for (int r = 0; r < DTRq; ++r) s += xr[r] * wr[r];
    dt[t] = (s > 20.f) ? s : log1pf(__expf(s));               // softplus
}

// ---------------------------------------------------------------------------
// Kernel 5: selective scan. Block = (g, b, 64-channel slice); thread = channel.
// h[16] in VGPRs; Bs/Cs staged into LDS via async B128 DMA (ASYNCcnt path),
// 64 timesteps per chunk. Fused: y = (sum_s h*C) + u*D, gated by silu(z).
// ---------------------------------------------------------------------------
#define SCAN_CH 64
#define SCAN_CL 64
__global__ __launch_bounds__(SCAN_CH) void scan_kernel(const float* __restrict__ xdbl,
                                                       const float* __restrict__ dt,
                                                       const float* __restrict__ u,
                                                       const float* __restrict__ zb,
                                                       const float* __restrict__ A_log,
                                                       const float* __restrict__ D_param,
                                                       float* __restrict__ y) {
    __shared__ float sBC[SCAN_CL * 32];     // per step: Bs[16] | Cs[16]
    const int nds = DIq / SCAN_CH;          // channel slices per (g,b)
    int blk    = blockIdx.x;
    int dchunk = blk % nds;
    int b      = (blk / nds) % Bq;
    int g      = blk / (nds * Bq);
    int d      = dchunk * SCAN_CH + threadIdx.x;
    int gd     = g * DIq + d;
    size_t rowbase = (size_t)g * BL + (size_t)b * Lq;   // (g,b) row at l=0

    float Anow[DSq];
#pragma unroll
    for (int s = 0; s < DSq; ++s) Anow[s] = -__expf(A_log[(size_t)gd * DSq + s]);
    float Dp = D_param[gd];
    float h[DSq];
#pragma unroll
    for (int s = 0; s < DSq; ++s) h[s] = 0.f;

    // wave-relative LDS byte offset of sBC (flat shared addr low 32 bits)
    unsigned ldsbase = (unsigned)(uintptr_t)(void*)sBC;

    for (int l0 = 0; l0 < Lq; l0 += SCAN_CL) {
        // Async DMA: 32 floats (Bs|Cs) per step = 8 x B128 per step,
        // SCAN_CL*8 = 512 transfers, 8 per thread. 16B aligned (row*160+32).
        for (int i = threadIdx.x; i < SCAN_CL * 8; i += SCAN_CH) {
            int st = i >> 3;      // step within chunk
            int q  = i & 7;       // 16B slice within the 128B Bs|Cs segment
            const float* gp = xdbl + (rowbase + l0 + st) * XDBL + DTRq + q * 4;
            unsigned loff = ldsbase + (unsigned)i * 16u;
            asm volatile("global_load_async_to_lds_b128 %0, %1, off"
                         :: "v"(loff), "v"(gp) : "memory");
        }
        asm volatile("s_wait_asynccnt 0x0" ::: "memory");
        __syncthreads();

        for (int l = l0; l < l0 + SCAN_CL; ++l) {
            size_t idx = (rowbase + l) * DIq + d;
            float dtv = dt[idx];
            float uv  = u[idx];
            const float* Bsh = &sBC[(l - l0) * 32];
            const float* Csh = Bsh + DSq;
            float acc = uv * Dp;
            float dBu = dtv * uv;
#pragma unroll
            for (int s = 0; s < DSq; ++s) {
                h[s] = __expf(dtv * Anow[s]) * h[s] + dBu * Bsh[s];
                acc += h[s] * Csh[s];
            }
            float zv = zb[idx];
            y[idx] = acc * silu_f(zv);
        }
        __syncthreads();
    }
}

// ---------------------------------------------------------------------------
// Kernel 6: out-projection. Per group: y_g[8192x256] * W_out[g]^T[256x128].
// One wave computes a 16(M) x 64(N) strip. Written straight into d_out.
// ---------------------------------------------------------------------------
__global__ __launch_bounds__(256) void outproj_kernel(const float* __restrict__ y,
                                                      const float* __restrict__ W_out,
                                                      float* __restrict__ out) {
    const int Mt = BL / 16;    // 512
    const int NW = GSq / 64;   // 2
    int wave = (blockIdx.x * blockDim.x + threadIdx.x) >> 5;
    int lane = threadIdx.x & 31;
    int nw = wave % NW;
    int mt = (wave / NW) % Mt;
    int g  = wave / (NW * Mt);
    if (g >= Gq) return;

    int mh = lane >> 4;
    int mn = lane & 15;
    const float* arow = y + ((size_t)g * BL + mt * 16 + mn) * DIq;
    const float* br[4];
#pragma unroll
    for (int j = 0; j < 4; ++j)
        br[j] = W_out + ((size_t)g * GSq + nw * 64 + j * 16 + mn) * DIq;

    v8f acc0 = {0.f,0.f,0.f,0.f,0.f,0.f,0.f,0.f};
    v8f acc1 = acc0, acc2 = acc0, acc3 = acc0;
#pragma unroll 4
    for (int k0 = 0; k0 < DIq; k0 += 4) {
        int kb = k0 + 2 * mh;
        v2f a  = *(const v2f*)(arow + kb);
        v2f b0 = *(const v2f*)(br[0] + kb);
        v2f b1 = *(const v2f*)(br[1] + kb);
        v2f b2 = *(const v2f*)(br[2] + kb);
        v2f b3 = *(const v2f*)(br[3] + kb);
        acc0 = wmma4(a, b0, acc0);
        acc1 = wmma4(a, b1, acc1);
        acc2 = wmma4(a, b2, acc2);
        acc3 = wmma4(a, b3, acc3);
    }
    v8f accs[4] = {acc0, acc1, acc2, acc3};
#pragma unroll
    for (int j = 0; j < 4; ++j) {
        int n = nw * 64 + j * 16 + mn;
#pragma unroll
        for (int r = 0; r < 8; ++r) {
            int orow = mt * 16 + r + 8 * mh;
            out[(size_t)orow * Dq + g * GSq + n] = accs[j][r];
        }
    }
}

// ---------------------------------------------------------------------------
extern "C" void kernel_launch(void* const* d_in, const int* in_sizes, int n_in,
                              void* d_out, int out_size, void* d_ws, size_t ws_size,
                              hipStream_t stream) {
    const float* x      = (const float*)d_in[0];
    const float* W_in   = (const float*)d_in[1];
    const float* conv_w = (const float*)d_in[2];
    const float* conv_b = (const float*)d_in[3];
    const float* W_x    = (const float*)d_in[4];
    const float* W_dt   = (const float*)d_in[5];
    const float* b_dt   = (const float*)d_in[6];
    const float* A_log  = (const float*)d_in[7];
    const float* D_par  = (const float*)d_in[8];
    const float* W_out  = (const float*)d_in[9];
    float* out = (float*)d_out;

    // Workspace carve (fp32): xc | z | u | dt | xdbl ; y reuses xc (dead after conv)
    const size_t NE = ROWS * DIq;           // 8.39M elements per big buffer
    float* xc   = (float*)d_ws;
    float* zb   = xc + NE;
    float* u    = zb + NE;
    float* dt   = u  + NE;
    float* xdbl = dt + NE;
    float* y    = xc;                       // reuse

    // 1) in-projection: 4 * 512 * 8 wave-strips, 8 waves/block
    {
        int waves = Gq * (BL / 16) * ((2 * DIq) / 64);   // 16384
        inproj_kernel<<<waves / 8, 256, 0, stream>>>(x, W_in, xc, zb);
    }
    // 2) depthwise conv + SiLU
    {
        size_t n = NE;
        conv_silu_kernel<<<(int)((n + 255) / 256), 256, 0, stream>>>(xc, conv_w, conv_b, u);
    }
    // 3) x_dbl projection (40 wide)
    {
        size_t n = ROWS * XDBL;
        xdbl_kernel<<<(int)((n + 255) / 256), 256, 0, stream>>>(u, W_x, xdbl);
    }
    // 4) dt = softplus(...)
    {
        size_t n = NE;
        dt_kernel<<<(int)((n + 255) / 256), 256, 0, stream>>>(xdbl, W_dt, b_dt, dt);
    }
    // 5) selective scan (sequential over L, parallel over g,b,channel slices)
    {
        int blocks = Gq * Bq * (DIq / SCAN_CH);   // 64
        scan_kernel<<<blocks, SCAN_CH, 0, stream>>>(xdbl, dt, u, zb, A_log, D_par, y);
    }
    // 6) out-projection straight into d_out
    {
        int waves = Gq * (BL / 16) * (GSq / 64);         // 4096
        outproj_kernel<<<waves / 8, 256, 0, stream>>>(y, W_out, out);
    }
}